// MockPlenoxels_44616120271621
// MI455X (gfx1250) — compile-verified
//
#include <hip/hip_runtime.h>

// ---------------------------------------------------------------------------
// MockPlenoxels: voxel gather + SH evaluation. Memory-bound (~0.7 FLOP/B).
// Strategy: wide 4B-aligned b128 gathers, global_prefetch_b8 to overlap the
// random SH-row gather with the SH-basis VALU work, non-temporal hints on
// streamed inputs/outputs so the 226MB sh_grid competes best for the 192MB L2.
// ---------------------------------------------------------------------------

typedef float f4a4 __attribute__((ext_vector_type(4), aligned(4)));

#define GRID_R 128
#define GRID_YZ (128 * 128)

__device__ __forceinline__ float grid_coord(float p) {
    // normalized = clip((p - (-1)) / 2, 0, 1); gc = clip(norm*128, 0, 127)
    float nrm = __fmul_rn(__fadd_rn(p, 1.0f), 0.5f);
    nrm = fminf(fmaxf(nrm, 0.0f), 1.0f);
    float g = __fmul_rn(nrm, 128.0f);
    return fminf(fmaxf(g, 0.0f), 127.0f);
}

__global__ __launch_bounds__(256) void plenoxels_kernel(
    const float* __restrict__ pos,      // [N,3]
    const float* __restrict__ dir,      // [N,3]
    const float* __restrict__ dens,     // [128^3]
    const float* __restrict__ sh,       // [128^3, 9, 3]
    float* __restrict__ out_density,    // [N]
    float* __restrict__ out_color,      // [N,3]
    int n)
{
    int i = blockIdx.x * blockDim.x + threadIdx.x;
    if (i >= n) return;

    // --- streamed position (read once -> non-temporal) ---
    float px = __builtin_nontemporal_load(pos + 3 * i + 0);
    float py = __builtin_nontemporal_load(pos + 3 * i + 1);
    float pz = __builtin_nontemporal_load(pos + 3 * i + 2);

    // --- voxel index, faithful to reference float math + single truncation ---
    float gx = grid_coord(px);
    float gy = grid_coord(py);
    float gz = grid_coord(pz);
    float fidx = __fadd_rn(__fadd_rn(__fmul_rn(gx, (float)GRID_YZ),
                                     __fmul_rn(gy, (float)GRID_R)), gz);
    int idx = (int)fidx;                       // >= 0, truncation toward zero

    const float* crow = sh + idx * 27;         // 108B contiguous, 4B aligned

    // --- gfx1250 prefetch: pull both cachelines of the SH row + density
    //     into the hierarchy before we burn VALU cycles on the basis ---
    __builtin_prefetch(crow, 0, 3);
    __builtin_prefetch(crow + 16, 0, 3);       // +64B: covers the 108B tail
    __builtin_prefetch(dens + idx, 0, 3);

    // --- streamed direction (read once -> non-temporal) ---
    float dx = __builtin_nontemporal_load(dir + 3 * i + 0);
    float dy = __builtin_nontemporal_load(dir + 3 * i + 1);
    float dz = __builtin_nontemporal_load(dir + 3 * i + 2);

    // --- issue the gathers early (temporal: want L2 residency for grids) ---
    float d = dens[idx];
    const f4a4* crow4 = (const f4a4*)crow;     // 4B-aligned b128 loads
    f4a4 v0 = crow4[0];
    f4a4 v1 = crow4[1];
    f4a4 v2 = crow4[2];
    f4a4 v3 = crow4[3];
    f4a4 v4 = crow4[4];
    f4a4 v5 = crow4[5];
    float c24 = crow[24], c25 = crow[25], c26 = crow[26];

    // --- SH basis (degree 2, 9 coeffs) — overlaps with gather latency ---
    const float k1 = 0.4886025119029199f;
    const float k2 = 1.0925484305920792f;
    float bs[9];
    bs[0] = 0.28209479177387814f;
    bs[1] = k1 * dy;
    bs[2] = k1 * dz;
    bs[3] = k1 * dx;
    bs[4] = k2 * dx * dy;
    bs[5] = k2 * dy * dz;
    bs[6] = 0.31539156525252005f * (3.0f * dz * dz - 1.0f);
    bs[7] = k2 * dx * dz;
    bs[8] = 0.5462742152960396f * (dx * dx - dy * dy);

    // --- unpack coeffs [9][3] and reduce ---
    float c[27];
    c[0]=v0.x;  c[1]=v0.y;  c[2]=v0.z;  c[3]=v0.w;
    c[4]=v1.x;  c[5]=v1.y;  c[6]=v1.z;  c[7]=v1.w;
    c[8]=v2.x;  c[9]=v2.y;  c[10]=v2.z; c[11]=v2.w;
    c[12]=v3.x; c[13]=v3.y; c[14]=v3.z; c[15]=v3.w;
    c[16]=v4.x; c[17]=v4.y; c[18]=v4.z; c[19]=v4.w;
    c[20]=v5.x; c[21]=v5.y; c[22]=v5.z; c[23]=v5.w;
    c[24]=c24;  c[25]=c25;  c[26]=c26;

    float cr = 0.0f, cg = 0.0f, cb = 0.0f;
    #pragma unroll
    for (int k = 0; k < 9; ++k) {
        cr = fmaf(bs[k], c[3 * k + 0], cr);
        cg = fmaf(bs[k], c[3 * k + 1], cg);
        cb = fmaf(bs[k], c[3 * k + 2], cb);
    }

    // --- activations ---
    float od = fmaxf(d, 0.0f);                       // relu
    float sr = 1.0f / (1.0f + expf(-cr));            // sigmoid
    float sg = 1.0f / (1.0f + expf(-cg));
    float sb = 1.0f / (1.0f + expf(-cb));

    // --- outputs: write-once -> non-temporal stores (protect L2) ---
    __builtin_nontemporal_store(od, out_density + i);
    __builtin_nontemporal_store(sr, out_color + 3 * i + 0);
    __builtin_nontemporal_store(sg, out_color + 3 * i + 1);
    __builtin_nontemporal_store(sb, out_color + 3 * i + 2);
}

extern "C" void kernel_launch(void* const* d_in, const int* in_sizes, int n_in,
                              void* d_out, int out_size, void* d_ws, size_t ws_size,
                              hipStream_t stream) {
    const float* positions  = (const float*)d_in[0];   // [N,3]
    const float* directions = (const float*)d_in[1];   // [N,3]
    const float* density    = (const float*)d_in[2];   // [128^3]
    const float* sh_grid    = (const float*)d_in[3];   // [128^3,9,3]

    const int n = in_sizes[0] / 3;                     // N points

    float* out_density = (float*)d_out;                // first N floats
    float* out_color   = (float*)d_out + n;            // then [N,3]

    dim3 block(256);                                   // 8 wave32 waves
    dim3 grid((n + 255) / 256);
    plenoxels_kernel<<<grid, block, 0, stream>>>(
        positions, directions, density, sh_grid, out_density, out_color, n);
}